// DiagonalSSMKernel_18769007084374
// MI455X (gfx1250) — compile-verified
//
#include <hip/hip_runtime.h>
#include <hip/hip_bf16.h>

typedef __attribute__((ext_vector_type(2))) float v2f;
typedef __attribute__((ext_vector_type(8))) float v8f;

#define NH    1024   // channels (D_MODEL)
#define NSTATE  32   // n = D_STATE/2
#define NL    2048   // sequence length
#define NDIR     2
#define ASTR    68   // LDS row stride (floats) for A panel, padded vs bank conflicts
#define BSTR   129   // LDS row stride (floats) for B panel

// One wave (32 lanes) per channel h. Lane index == state index n.
// Factorization: r^l = (r^16)^i * r^j with l = 16*i + j.
//   A[(d*16+j)][2n+0] =  Re(sc[d,n] * r^j)
//   A[(d*16+j)][2n+1] = -Im(sc[d,n] * r^j)
//   B[2n+0][i]        =  Re((r^16)^i)
//   B[2n+1][i]        =  Im((r^16)^i)
// out[d,h,16i+j] = 2 * (A @ B)[(d*16+j)][i]   -- 32x64x128 f32 GEMM via
// v_wmma_f32_16x16x4_f32 (2 M-tiles x 8 N-tiles x 16 K-steps).
__global__ __launch_bounds__(32) void ssm_vandermonde_wmma(
    const float* __restrict__ log_dt,
    const float* __restrict__ log_a_real,
    const float* __restrict__ a_imag,
    const float* __restrict__ coeffs,
    float* __restrict__ out)
{
  __shared__ float Asm[32 * ASTR];   //  8.5 KB
  __shared__ float Bsm[64 * BSTR];   // 33.0 KB

  const int h    = blockIdx.x;
  const int lane = threadIdx.x;      // = n, 0..31

  // ---- per-state setup (few transcendentals per lane) ----
  const float dt  = __expf(log_dt[h]);
  const float ar  = -__expf(log_a_real[h * NSTATE + lane]);
  const float ai  = a_imag[h * NSTATE + lane];
  const float dar = ar * dt;
  const float dai = ai * dt;
  const float e   = __expf(dar);
  const float rr  = e * __cosf(dai);          // r = exp(delta_a)
  const float ri  = e * __sinf(dai);
  // w = (exp(delta_a) - 1) / a   (complex divide by a = ar + i*ai)
  const float nr  = rr - 1.0f, ni = ri;
  const float inv = 1.0f / (ar * ar + ai * ai);
  const float wr  = (nr * ar + ni * ai) * inv;
  const float wi  = (ni * ar - nr * ai) * inv;
  // sc_d = c_d * w ; coeffs layout (D, H, N, 2) interleaved re/im
  const size_t cbase = ((size_t)h * NSTATE + lane) * 2;
  const float c0r = coeffs[cbase + 0];
  const float c0i = coeffs[cbase + 1];
  const float c1r = coeffs[(size_t)NH * NSTATE * 2 + cbase + 0];
  const float c1i = coeffs[(size_t)NH * NSTATE * 2 + cbase + 1];
  const float s0r = c0r * wr - c0i * wi, s0i = c0r * wi + c0i * wr;
  const float s1r = c1r * wr - c1i * wi, s1i = c1r * wi + c1i * wr;

  // ---- A panel: T[d,n,j] = sc_d * r^j  (recurrence over j) ----
  float br = 1.0f, bi = 0.0f;                 // r^j
  #pragma unroll
  for (int j = 0; j < 16; ++j) {
    const float t0r = s0r * br - s0i * bi, t0i = s0r * bi + s0i * br;
    const float t1r = s1r * br - s1i * bi, t1i = s1r * bi + s1i * br;
    Asm[( 0 + j) * ASTR + 2 * lane + 0] =  t0r;
    Asm[( 0 + j) * ASTR + 2 * lane + 1] = -t0i;
    Asm[(16 + j) * ASTR + 2 * lane + 0] =  t1r;
    Asm[(16 + j) * ASTR + 2 * lane + 1] = -t1i;
    const float nbr = br * rr - bi * ri;
    bi = br * ri + bi * rr;
    br = nbr;
  }
  // after 16 iterations (br,bi) = r^16
  const float prr = br, pri = bi;

  // ---- B panel: step[n,i] = (r^16)^i  (recurrence over i) ----
  float sr = 1.0f, si = 0.0f;
  #pragma unroll 4
  for (int i = 0; i < 128; ++i) {
    Bsm[(2 * lane + 0) * BSTR + i] = sr;
    Bsm[(2 * lane + 1) * BSTR + i] = si;
    const float nsr = sr * prr - si * pri;
    si = sr * pri + si * prr;
    sr = nsr;
  }

  __syncthreads();  // single wave: just orders LDS writes vs reads

  // ---- 32x64x128 GEMM with V_WMMA_F32_16X16X4_F32 ----
  const int lm = lane & 15;         // M / N position within tile
  const int lh = lane >> 4;         // K-half selector (0 or 1)

  #pragma unroll
  for (int mt = 0; mt < 2; ++mt) {          // mt == direction d
    #pragma unroll
    for (int nt = 0; nt < 8; ++nt) {        // i-tile
      v8f acc = {};
      #pragma unroll
      for (int kt = 0; kt < 16; ++kt) {     // K in steps of 4
        const int kb2 = kt * 4 + lh * 2;
        v2f a, b;
        // A 16x4 frag: lanes 0-15 -> K = kb,kb+1 ; lanes 16-31 -> K = kb+2,kb+3
        a.x = Asm[(mt * 16 + lm) * ASTR + kb2 + 0];
        a.y = Asm[(mt * 16 + lm) * ASTR + kb2 + 1];
        // B 4x16 frag: VGPR0 rows {kb, kb+2}, VGPR1 rows {kb+1, kb+3}
        b.x = Bsm[(kb2 + 0) * BSTR + nt * 16 + lm];
        b.y = Bsm[(kb2 + 1) * BSTR + nt * 16 + lm];
        acc = __builtin_amdgcn_wmma_f32_16x16x4_f32(
            /*neg_a=*/false, a, /*neg_b=*/false, b,
            /*c_mod=*/(short)0, acc, /*reuse_a=*/false, /*reuse_b=*/false);
      }
      // C/D layout: VGPR v -> row (v + 8*lh); col = lm
      // row = d*16 + j  => j = v + 8*lh ;  col -> i = nt*16 + lm
      const int i_col = nt * 16 + lm;
      float* outp = out + ((size_t)mt * NH + h) * NL + (size_t)i_col * 16;
      #pragma unroll
      for (int v = 0; v < 8; ++v) {
        outp[v + 8 * lh] = 2.0f * acc[v];
      }
    }
  }
}

extern "C" void kernel_launch(void* const* d_in, const int* in_sizes, int n_in,
                              void* d_out, int out_size, void* d_ws, size_t ws_size,
                              hipStream_t stream) {
  (void)in_sizes; (void)n_in; (void)d_ws; (void)ws_size; (void)out_size;
  const float* log_dt     = (const float*)d_in[0];   // (1024,)
  const float* log_a_real = (const float*)d_in[1];   // (1024, 32)
  const float* a_imag     = (const float*)d_in[2];   // (1024, 32)
  const float* coeffs     = (const float*)d_in[3];   // (2, 1024, 32, 2)
  // d_in[4] = sequence_length (==2048, compile-time constant here)
  float* out = (float*)d_out;                        // (2, 1024, 2048)

  ssm_vandermonde_wmma<<<NH, 32, 0, stream>>>(log_dt, log_a_real, a_imag,
                                              coeffs, out);
}